// KANEdgeFunction_51539607552785
// MI455X (gfx1250) — compile-verified
//
#include <hip/hip_runtime.h>

// ---------------------------------------------------------------------------
// KAN edge function: out = w_base*silu(x) + w_spline * dot(B(x), coeffs)
// Cubic B-spline, 64-point clamped uniform grid on [-5,5], 66 basis funcs.
// Streaming elementwise kernel; CDNA5 async global->LDS double buffering.
// ---------------------------------------------------------------------------

#define THREADS 256
#define F4_PER_THREAD 2
#define TILE_F4 (THREADS * F4_PER_THREAD) // 512 float4 = 8KB per tile

typedef int v4i __attribute__((ext_vector_type(4)));

static __device__ __forceinline__ void async_copy_f4(const float4* gsrc, float4* ldst) {
#if __has_builtin(__builtin_amdgcn_global_load_async_to_lds_b128)
  __builtin_amdgcn_global_load_async_to_lds_b128(
      (v4i*)gsrc, (v4i*)ldst, /*offset=*/0, /*cpol=*/0);
#else
  unsigned l32 = (unsigned)(unsigned long long)ldst;
  unsigned long long g64 = (unsigned long long)gsrc;
  asm volatile("global_load_async_to_lds_b128 %0, %1, off"
               :: "v"(l32), "v"(g64) : "memory");
#endif
}

template <int N>
static __device__ __forceinline__ void wait_asynccnt() {
#if __has_builtin(__builtin_amdgcn_s_wait_asynccnt)
  __builtin_amdgcn_s_wait_asynccnt(N);
#else
  asm volatile("s_wait_asynccnt %0" :: "i"(N) : "memory");
#endif
}

// Evaluate one element. sc points to coeffs staged in LDS.
static __device__ __forceinline__ float kan_eval(float x, const float* sc,
                                                 float wbv, float wsv) {
  // ---- SiLU: x * sigmoid(x) ----
  const float LOG2E = 1.44269504088896340736f;
  float sig  = __builtin_amdgcn_rcpf(1.0f + __builtin_amdgcn_exp2f(-x * LOG2E));
  float base = x * sig;

  // ---- span location: grid cell k in [0,62], h = 10/63 ----
  const float H    = 0.15873015873015872f; // 10/63
  const float INVH = 6.3f;                 // 63/10
  float kf = floorf((x + 5.0f) * INVH);
  kf = __builtin_amdgcn_fmed3f(kf, 0.0f, 62.0f); // clamp (also handles OOB x)
  int k = (int)kf;

  // ---- knots: full_knots[m] = -5 + H*clamp(m-3, 0, 63); need m=k+1..k+6 ----
  float c;
  c = __builtin_amdgcn_fmed3f(kf - 2.0f, 0.0f, 63.0f); float tA = fmaf(c, H, -5.0f); // T[k+1]
  c = __builtin_amdgcn_fmed3f(kf - 1.0f, 0.0f, 63.0f); float tB = fmaf(c, H, -5.0f); // T[k+2]
  c = __builtin_amdgcn_fmed3f(kf       , 0.0f, 63.0f); float tC = fmaf(c, H, -5.0f); // T[k+3]
  c = __builtin_amdgcn_fmed3f(kf + 1.0f, 0.0f, 63.0f); float tD = fmaf(c, H, -5.0f); // T[k+4]
  c = __builtin_amdgcn_fmed3f(kf + 2.0f, 0.0f, 63.0f); float tE = fmaf(c, H, -5.0f); // T[k+5]
  c = __builtin_amdgcn_fmed3f(kf + 3.0f, 0.0f, 63.0f); float tF = fmaf(c, H, -5.0f); // T[k+6]

  float l1 = x - tC, l2 = x - tB, l3 = x - tA;
  float r1 = tD - x, r2 = tE - x, r3 = tF - x;

  // ---- de Boor (Cox-de Boor restricted to the 4 active cubic bases) ----
  float temp, saved;
  // degree 1
  temp = __builtin_amdgcn_rcpf(r1 + l1);
  float N0 = r1 * temp;
  float N1 = l1 * temp;
  // degree 2
  temp = N0 * __builtin_amdgcn_rcpf(r1 + l2);
  N0 = r1 * temp;              saved = l2 * temp;
  temp = N1 * __builtin_amdgcn_rcpf(r2 + l1);
  N1 = fmaf(r2, temp, saved);  float N2 = l1 * temp;
  // degree 3
  temp = N0 * __builtin_amdgcn_rcpf(r1 + l3);
  N0 = r1 * temp;              saved = l3 * temp;
  temp = N1 * __builtin_amdgcn_rcpf(r2 + l2);
  N1 = fmaf(r2, temp, saved);  saved = l2 * temp;
  temp = N2 * __builtin_amdgcn_rcpf(r3 + l1);
  N2 = fmaf(r3, temp, saved);  float N3 = l1 * temp;

  // ---- dot with the 4 active coefficients (LDS) ----
  float spline = N0 * sc[k];
  spline = fmaf(N1, sc[k + 1], spline);
  spline = fmaf(N2, sc[k + 2], spline);
  spline = fmaf(N3, sc[k + 3], spline);
  bool valid = (x >= -5.0f) && (x < 5.0f); // basis is 0 outside [-5,5)
  spline = valid ? spline : 0.0f;

  return fmaf(wbv, base, wsv * spline);
}

__global__ __launch_bounds__(THREADS)
void kan_bspline_kernel(const float* __restrict__ x,
                        const float* __restrict__ w_base,
                        const float* __restrict__ w_spline,
                        const float* __restrict__ coeffs,
                        float* __restrict__ out,
                        int n, int n4, int num_tiles) {
  __shared__ float4 buf[2][TILE_F4]; // 16 KB double buffer
  __shared__ float  scoef[68];

  const int tid = threadIdx.x;
  if (tid < 66) scoef[tid] = coeffs[tid];
  const float wbv = w_base[0];   // uniform -> scalar loads
  const float wsv = w_spline[0];

  const float4* x4   = (const float4*)x;
  float4*       out4 = (float4*)out;

  // ---- preload first tile asynchronously ----
  const int t0 = blockIdx.x;
  if (t0 < num_tiles) {
#pragma unroll
    for (int j = 0; j < F4_PER_THREAD; ++j) {
      int local = j * THREADS + tid;
      int idx   = t0 * TILE_F4 + local;
      idx = (idx < n4) ? idx : (n4 - 1); // clamp; store is guarded later
      async_copy_f4(x4 + idx, &buf[0][local]);
    }
  }
  __syncthreads(); // coeffs visible to all waves

  int cur = 0;
  for (int t = t0; t < num_tiles; t += (int)gridDim.x) {
    const int tn = t + (int)gridDim.x;
    if (tn < num_tiles) {
      // prefetch next tile into the other buffer while tile t is consumed
#pragma unroll
      for (int j = 0; j < F4_PER_THREAD; ++j) {
        int local = j * THREADS + tid;
        int idx   = tn * TILE_F4 + local;
        idx = (idx < n4) ? idx : (n4 - 1);
        async_copy_f4(x4 + idx, &buf[cur ^ 1][local]);
      }
      wait_asynccnt<F4_PER_THREAD>(); // oldest (tile t) async ops done
    } else {
      wait_asynccnt<0>();
    }
    __syncthreads(); // all waves' tile-t loads are visible in LDS

#pragma unroll
    for (int j = 0; j < F4_PER_THREAD; ++j) {
      int local = j * THREADS + tid;
      int idx   = t * TILE_F4 + local;
      float4 v = buf[cur][local];
      float4 r;
      r.x = kan_eval(v.x, scoef, wbv, wsv);
      r.y = kan_eval(v.y, scoef, wbv, wsv);
      r.z = kan_eval(v.z, scoef, wbv, wsv);
      r.w = kan_eval(v.w, scoef, wbv, wsv);
      if (idx < n4) out4[idx] = r;
    }
    __syncthreads(); // everyone done reading buf[cur] before it is refilled
    cur ^= 1;
  }

  // ---- scalar tail if n not a multiple of 4 ----
  const int rem = n - (n4 << 2);
  if (blockIdx.x == 0 && tid < rem) {
    float xv = x[(n4 << 2) + tid];
    out[(n4 << 2) + tid] = kan_eval(xv, scoef, wbv, wsv);
  }
}

extern "C" void kernel_launch(void* const* d_in, const int* in_sizes, int n_in,
                              void* d_out, int out_size, void* d_ws, size_t ws_size,
                              hipStream_t stream) {
  const float* x     = (const float*)d_in[0];
  const float* wb    = (const float*)d_in[1];
  const float* ws    = (const float*)d_in[2];
  const float* coefs = (const float*)d_in[3];
  float* out = (float*)d_out;

  const int n  = in_sizes[0];
  const int n4 = n >> 2;
  const int num_tiles = (n4 + TILE_F4 - 1) / TILE_F4;

  int grid = num_tiles > 1024 ? 1024 : num_tiles;
  if (grid < 1) grid = 1; // tail-only case

  kan_bspline_kernel<<<grid, THREADS, 0, stream>>>(x, wb, ws, coefs, out,
                                                   n, n4, num_tiles);
}